// partical_similarity_70446053589133
// MI455X (gfx1250) — compile-verified
//
#include <hip/hip_runtime.h>
#include <hip/hip_bf16.h>

typedef __attribute__((ext_vector_type(2))) float v2f;
typedef __attribute__((ext_vector_type(8))) float v8f;

#define NCH   64      // channels C
#define NM    64      // sat batch M
#define NN    64      // grd batch N (filters)
#define HS    64      // sat_feature H/W
#define KH    24
#define KW    24
#define HO    17
#define WO    17
#define NP    (HO*WO)         // 289 shift positions
#define KC    192             // K-chunk: 1 channel x 8 rows x 24
#define ASTR  196             // LDS row stride (floats): 16B-aligned rows, conflict-free
#define BSTR  196
#define KTOT  (NCH*KH*KW)     // 36864
#define EPSF  1e-12f

// ---------------- corr GEMM: f32 WMMA 16x16x4 ----------------
// grid = (64 m, 10 ptiles of 32 positions), block = 256 (8 waves).
// Wave w computes position-half (w>>2) x filter-tile (w&3): a 16x16 f32 tile.
__global__ __launch_bounds__(256) void corr_kernel(const float* __restrict__ grd,
                                                   const float* __restrict__ sat,
                                                   float* __restrict__ corr) {
  __shared__ float As[32 * ASTR];   // 32 positions x 192 k-elems
  __shared__ float Bs[NN * BSTR];   // 64 filters  x 192 k-elems (k-contiguous)

  const int m     = blockIdx.x;
  const int ptile = blockIdx.y;
  const int tid   = threadIdx.x;
  const int wave  = tid >> 5;       // wave32
  const int lane  = tid & 31;

  const float* satm = sat + (size_t)m * NCH * HS * HS;

  // --- A staging map: 8 threads per position, one kernel-row (24 floats) each ---
  const int ai  = tid >> 3;                  // position slot 0..31
  const int ah0 = tid & 7;                   // kernel row within chunk 0..7
  int p = ptile * 32 + ai; if (p > NP - 1) p = NP - 1;   // clamp pad rows
  const int ay = p / WO, ax = p - (p / WO) * WO;
  const int abase = (12 + ay + ah0) * HS + (12 + ax);    // per-channel offset, hb=0
  float* aDst = &As[ai * ASTR + ah0 * KW];

  // --- B staging map: 4 threads per filter, 48 contiguous k-elems each ---
  const int bn  = tid >> 2;                  // filter 0..63
  const int bk0 = (tid & 3) * 48;
  const float* bSrc0 = grd + (size_t)bn * KTOT + bk0;    // + c*576 + hb*24
  float4* bDst = (float4*)&Bs[bn * BSTR + bk0];

  // --- WMMA fragment pointers (ISA 7.12.2 32-bit layouts) ---
  const int row  = (lane & 15) + ((wave >> 2) << 4);     // A row = position
  const int koff = (lane >> 4) << 1;                     // 0 or 2
  const float* ap = &As[row * ASTR + koff];
  const float* bp = &Bs[(((wave & 3) << 4) + (lane & 15)) * BSTR + koff];

  v8f acc = {};

  for (int c = 0; c < NCH; ++c) {
    const float* satc = satm + (size_t)c * HS * HS;
    const float* gC   = bSrc0 + (size_t)c * (KH * KW);
    for (int hb = 0; hb < KH; hb += 8) {
      // ---- stage A: 24 contiguous floats from one sat row ----
      const float* aSrc = satc + abase + hb * HS;
#pragma unroll
      for (int j = 0; j < KW; ++j) aDst[j] = aSrc[j];
      // ---- stage B: 48 contiguous floats as 12 x float4 ----
      const float4* bSrc = (const float4*)(gC + hb * KW);
#pragma unroll
      for (int j = 0; j < 12; ++j) bDst[j] = bSrc[j];
      __syncthreads();

      // warm caches for the next chunk while the matrix pipe works
      __builtin_prefetch(aSrc + 8 * HS, 0, 0);
      if (c < NCH - 1 || hb < 16) __builtin_prefetch(gC + (hb + 8) * KW, 0, 0);

#pragma unroll
      for (int kt = 0; kt < KC; kt += 4) {
        v2f a = *(const v2f*)(ap + kt);     // A[M=row][k0+koff .. +1]
        v2f b = *(const v2f*)(bp + kt);     // B[k0+koff .. +1][N]
        acc = __builtin_amdgcn_wmma_f32_16x16x4_f32(false, a, false, b,
                                                    (short)0, acc, false, false);
      }
      __syncthreads();
    }
  }

  // C/D layout: VGPR v, lanes 0-15 -> M=v, lanes 16-31 -> M=v+8; N = lane&15
  const int n     = ((wave & 3) << 4) + (lane & 15);
  const int ibase = ((wave >> 2) << 4) + ((lane >> 4) << 3);
#pragma unroll
  for (int v = 0; v < 8; ++v) {
    int pp = ptile * 32 + ibase + v;
    if (pp < NP) corr[((size_t)m * NN + n) * NP + pp] = acc[v];
  }
}

// ---------------- argmax over 289 positions per (m,n) ----------------
__global__ __launch_bounds__(32) void argmax_kernel(const float* __restrict__ corr,
                                                    float* __restrict__ maxval,
                                                    int* __restrict__ maxidx) {
  const int mn = blockIdx.x;
  const float* rowp = corr + (size_t)mn * NP;
  const int lane = threadIdx.x;
  float best = -3.402823466e38f;
  int   bidx = 0x7fffffff;
  for (int p = lane; p < NP; p += 32) {
    float v = rowp[p];
    if (v > best) { best = v; bidx = p; }
  }
  for (int off = 16; off > 0; off >>= 1) {
    float ov = __shfl_down(best, off, 32);
    int   oi = __shfl_down(bidx, off, 32);
    if (ov > best || (ov == best && oi < bidx)) { best = ov; bidx = oi; }
  }
  if (lane == 0) { maxval[mn] = best; maxidx[mn] = bidx; }
}

// ---------------- per-filter grd norms ----------------
__global__ __launch_bounds__(256) void gnorm_kernel(const float* __restrict__ grd,
                                                    float* __restrict__ gnorm) {
  const int n = blockIdx.x;
  const float* g = grd + (size_t)n * KTOT;
  float s = 0.f;
  for (int e = threadIdx.x; e < KTOT; e += 256) { float v = g[e]; s += v * v; }
  __shared__ float red[8];
  for (int off = 16; off > 0; off >>= 1) s += __shfl_down(s, off, 32);
  if ((threadIdx.x & 31) == 0) red[threadIdx.x >> 5] = s;
  __syncthreads();
  if (threadIdx.x < 8) {
    s = red[threadIdx.x];
    for (int off = 4; off > 0; off >>= 1) s += __shfl_down(s, off, 8);
    if (threadIdx.x == 0) gnorm[n] = fmaxf(sqrtf(s), EPSF);
  }
}

// ---------------- patch norm @ argmax + similarity ----------------
__global__ __launch_bounds__(256) void sim_kernel(const float* __restrict__ sat,
                                                  const float* __restrict__ maxval,
                                                  const int* __restrict__ maxidx,
                                                  const float* __restrict__ gnorm,
                                                  float* __restrict__ out) {
  const int mn = blockIdx.x;
  const int m = mn >> 6, n = mn & 63;
  const int idx = maxidx[mn];
  const int hy = idx / WO, wx = idx % WO;
  const float* sm = sat + (size_t)m * NCH * HS * HS;
  float s = 0.f;
  for (int e = threadIdx.x; e < KTOT; e += 256) {
    int c = e / (KH * KW), r = e - c * (KH * KW);
    int h = r / KW, w = r - h * KW;
    float v = sm[((size_t)c * HS + (12 + hy + h)) * HS + (12 + wx + w)];
    s += v * v;
  }
  __shared__ float red[8];
  for (int off = 16; off > 0; off >>= 1) s += __shfl_down(s, off, 32);
  if ((threadIdx.x & 31) == 0) red[threadIdx.x >> 5] = s;
  __syncthreads();
  if (threadIdx.x < 8) {
    s = red[threadIdx.x];
    for (int off = 4; off > 0; off >>= 1) s += __shfl_down(s, off, 8);
    if (threadIdx.x == 0)
      out[mn] = maxval[mn] / (fmaxf(sqrtf(s), EPSF) * gnorm[n]);
  }
}

// ---------------- sat_f = patch at (m=63, n=63) ----------------
__global__ __launch_bounds__(256) void satf_kernel(const float* __restrict__ sat,
                                                   const int* __restrict__ maxidx,
                                                   float* __restrict__ out) {
  const int idx = maxidx[NM * NN - 1];
  const int hy = idx / WO, wx = idx % WO;
  const float* sm = sat + (size_t)(NM - 1) * NCH * HS * HS;
  int e = blockIdx.x * 256 + threadIdx.x;
  if (e < KTOT) {
    int c = e / (KH * KW), r = e - c * (KH * KW);
    int h = r / KW, w = r - h * KW;
    out[NM * NN + e] = sm[((size_t)c * HS + (12 + hy + h)) * HS + (12 + wx + w)];
  }
}

extern "C" void kernel_launch(void* const* d_in, const int* in_sizes, int n_in,
                              void* d_out, int out_size, void* d_ws, size_t ws_size,
                              hipStream_t stream) {
  const float* grd = (const float*)d_in[0];   // (64,64,24,24)
  const float* sat = (const float*)d_in[1];   // (64,64,64,64)
  float* out = (float*)d_out;                 // 4096 sims + 36864 sat_f

  char* ws = (char*)d_ws;
  const size_t corrBytes = (size_t)NM * NN * NP * sizeof(float);   // ~4.73 MB
  float* corr   = (float*)ws;
  float* maxval = (float*)(ws + corrBytes);
  int*   maxidx = (int*)  (ws + corrBytes + 16384);
  float* gnorm  = (float*)(ws + corrBytes + 32768);

  corr_kernel  <<<dim3(NM, (NP + 31) / 32), 256, 0, stream>>>(grd, sat, corr);
  argmax_kernel<<<NM * NN, 32, 0, stream>>>(corr, maxval, maxidx);
  gnorm_kernel <<<NN, 256, 0, stream>>>(grd, gnorm);
  sim_kernel   <<<NM * NN, 256, 0, stream>>>(sat, maxval, maxidx, gnorm, out);
  satf_kernel  <<<(KTOT + 255) / 256, 256, 0, stream>>>(sat, maxidx, out);
}